// Sampler_18416819765664
// MI455X (gfx1250) — compile-verified
//
#include <hip/hip_runtime.h>
#include <cstdint>
#include <cstddef>

typedef float vfloat4 __attribute__((ext_vector_type(4)));

namespace {
constexpr int   kVocab   = 2048;
constexpr int   kThreads = 256;                  // 8 wave32 per row
constexpr int   kWaves   = kThreads / 32;
constexpr int   kPerThr  = kVocab / kThreads;    // 8 elements / thread
constexpr int   kBins    = 2048;                 // value-histogram bins
constexpr int   kBinsPT  = kBins / kThreads;     // 8 bins / thread
constexpr float kInvW    = (float)kBins / 32.0f; // bins over [max-32, max]
constexpr float kTopP    = 0.95f;
constexpr float kInvTemp = 1.0f / 0.8f;
}

__global__ __launch_bounds__(kThreads)
void topp_sampler_kernel(const float* __restrict__ logits,
                         const float* __restrict__ uni,
                         float* __restrict__ out_sampled,
                         float* __restrict__ out_probs)
{
    __shared__ float sh_row[kVocab];      // 8 KB: one logits row
    __shared__ float sh_hist[kBins];      // 8 KB: exp-mass histogram by value
    __shared__ float sh_wmax[kWaves];
    __shared__ float sh_wsum[kWaves];
    __shared__ float sh_wscan[kWaves];
    __shared__ float sh_wbest[kWaves];
    __shared__ int   sh_wbidx[kWaves];
    __shared__ int   sh_cbin;
    __shared__ float sh_zk;

    const int    tid   = threadIdx.x;
    const int    lane  = tid & 31;        // wave32 lane
    const int    wave  = tid >> 5;
    const int    row   = blockIdx.x;
    const size_t rbase = (size_t)row * kVocab;

    // ---- CDNA5 async global->LDS copy of the logits row (ASYNCcnt path) ----
    {
        const float* gp   = logits + rbase + tid * 4;            // 16 B / lane / issue
        uint32_t     loff = (uint32_t)(uintptr_t)&sh_row[tid * 4];
        asm volatile(
            "global_load_async_to_lds_b128 %0, %1, off\n\t"
            "global_load_async_to_lds_b128 %0, %1, off offset:4096"
            :: "v"(loff), "v"(gp)
            : "memory");
    }

    // Zero the histogram while the async copy is in flight.
    #pragma unroll
    for (int k = 0; k < kBinsPT; ++k)
        sh_hist[tid + k * kThreads] = 0.0f;

    asm volatile("s_wait_asynccnt 0" ::: "memory");
    __syncthreads();                                             // barrier 1

    // ---- pass 1: temperature scale + row max (wave shuffle reduce) ----
    float l[kPerThr];
    {
        const vfloat4 r0 = *reinterpret_cast<const vfloat4*>(&sh_row[tid * kPerThr]);
        const vfloat4 r1 = *reinterpret_cast<const vfloat4*>(&sh_row[tid * kPerThr + 4]);
        l[0] = r0.x * kInvTemp; l[1] = r0.y * kInvTemp;
        l[2] = r0.z * kInvTemp; l[3] = r0.w * kInvTemp;
        l[4] = r1.x * kInvTemp; l[5] = r1.y * kInvTemp;
        l[6] = r1.z * kInvTemp; l[7] = r1.w * kInvTemp;
    }
    float lmax = l[0];
    #pragma unroll
    for (int k = 1; k < kPerThr; ++k) lmax = fmaxf(lmax, l[k]);
    #pragma unroll
    for (int m = 16; m > 0; m >>= 1)
        lmax = fmaxf(lmax, __shfl_xor(lmax, m, 32));
    if (lane == 0) sh_wmax[wave] = lmax;
    __syncthreads();                                             // barrier 2
    float rmax = sh_wmax[0];
    #pragma unroll
    for (int w = 1; w < kWaves; ++w) rmax = fmaxf(rmax, sh_wmax[w]);

    // ---- pass 2: exp, Z, and exp-mass histogram (ds_add_f32 atomics) ----
    float e[kPerThr];
    int   bin[kPerThr];
    float zsum = 0.0f;
    #pragma unroll
    for (int k = 0; k < kPerThr; ++k) {
        e[k] = __expf(l[k] - rmax);
        zsum += e[k];
        int b = (int)((rmax - l[k]) * kInvW);      // 0 = highest-value bin
        b = b < 0 ? 0 : (b > kBins - 1 ? kBins - 1 : b);
        bin[k] = b;
        atomicAdd(&sh_hist[b], e[k]);
    }
    #pragma unroll
    for (int m = 16; m > 0; m >>= 1)
        zsum += __shfl_xor(zsum, m, 32);
    if (lane == 0) sh_wsum[wave] = zsum;
    __syncthreads();                                             // barrier 3
    float Z = 0.0f;
    #pragma unroll
    for (int w = 0; w < kWaves; ++w) Z += sh_wsum[w];
    const float target = kTopP * Z;

    // ---- value-ordered prefix sum over bins (wave shuffle scan) ----
    float bs = 0.0f;
    #pragma unroll
    for (int k = 0; k < kBinsPT; ++k) bs += sh_hist[tid * kBinsPT + k];
    float sc = bs;                                   // intra-wave inclusive scan
    #pragma unroll
    for (int off = 1; off < 32; off <<= 1) {
        const float t = __shfl_up(sc, off, 32);
        if (lane >= off) sc += t;
    }
    if (lane == 31) sh_wscan[wave] = sc;             // wave totals
    if (tid == 0)   sh_cbin = -1;
    __syncthreads();                                             // barrier 4
    float woff = 0.0f, total = 0.0f;
    #pragma unroll
    for (int w = 0; w < kWaves; ++w) {
        const float t = sh_wscan[w];
        if (w < wave) woff += t;
        total += t;
    }
    const float incl = woff + sc;
    const float excl = incl - bs;
    if (excl <= target && incl > target) {           // unique crossing thread
        float run = excl;
        int   c   = tid * kBinsPT;
        #pragma unroll
        for (int k = 0; k < kBinsPT; ++k) {
            run += sh_hist[tid * kBinsPT + k];
            if (run > target) { c = tid * kBinsPT + k; break; }
        }
        sh_cbin = c;       // first bin crossing p*Z is kept (reference shift)
        sh_zk   = run;     // kept-set mass = softmax renormalizer
    }
    __syncthreads();                                             // barrier 5

    int   cbin = sh_cbin;
    float zk   = (cbin < 0) ? total : sh_zk;         // fallback: keep everything
    if (cbin < 0) cbin = kBins - 1;
    const float invzk = 1.0f / zk;

    // ---- output pass: renormalized probs + fused Gumbel-max (streaming NT) ----
    const vfloat4 u0 = __builtin_nontemporal_load(
        reinterpret_cast<const vfloat4*>(uni + rbase + tid * kPerThr));
    const vfloat4 u1 = __builtin_nontemporal_load(
        reinterpret_cast<const vfloat4*>(uni + rbase + tid * kPerThr + 4));
    const float uu[kPerThr] = {u0.x, u0.y, u0.z, u0.w, u1.x, u1.y, u1.z, u1.w};

    float pout[kPerThr];
    float best    = -__builtin_inff();
    int   bestIdx = 0;
    #pragma unroll
    for (int k = 0; k < kPerThr; ++k) {
        const bool  keep = (bin[k] <= cbin);
        pout[k] = keep ? e[k] * invzk : 0.0f;
        const float g  = -__logf(-__logf(uu[k]));               // Gumbel noise
        const float sv = keep ? (l[k] + g) : -__builtin_inff();
        if (sv > best) { best = sv; bestIdx = tid * kPerThr + k; }
    }
    const vfloat4 p0 = {pout[0], pout[1], pout[2], pout[3]};
    const vfloat4 p1 = {pout[4], pout[5], pout[6], pout[7]};
    __builtin_nontemporal_store(p0,
        reinterpret_cast<vfloat4*>(out_probs + rbase + tid * kPerThr));
    __builtin_nontemporal_store(p1,
        reinterpret_cast<vfloat4*>(out_probs + rbase + tid * kPerThr + 4));

    // ---- argmax: wave shuffle butterfly, then 8-entry cross-wave reduce ----
    #pragma unroll
    for (int m = 16; m > 0; m >>= 1) {
        const float ov = __shfl_xor(best, m, 32);
        const int   oi = __shfl_xor(bestIdx, m, 32);
        if (ov > best || (ov == best && oi < bestIdx)) { best = ov; bestIdx = oi; }
    }
    if (lane == 0) { sh_wbest[wave] = best; sh_wbidx[wave] = bestIdx; }
    __syncthreads();                                             // barrier 6
    if (tid == 0) {
        float bv = sh_wbest[0];
        int   bi = sh_wbidx[0];
        #pragma unroll
        for (int w = 1; w < kWaves; ++w) {
            const float ov = sh_wbest[w];
            const int   oi = sh_wbidx[w];
            if (ov > bv || (ov == bv && oi < bi)) { bv = ov; bi = oi; }
        }
        out_sampled[row] = (float)bi;
    }
}

extern "C" void kernel_launch(void* const* d_in, const int* in_sizes, int n_in,
                              void* d_out, int out_size, void* d_ws, size_t ws_size,
                              hipStream_t stream) {
    const float* logits = (const float*)d_in[0];
    const float* uni    = (const float*)d_in[1];
    const int B = in_sizes[0] / kVocab;

    float* out_sampled = (float*)d_out;       // first B entries: sampled indices
    float* out_probs   = (float*)d_out + B;   // then B*V renormalized probs

    topp_sampler_kernel<<<B, kThreads, 0, stream>>>(logits, uni, out_sampled, out_probs);
}